// OpenBoundary_19129784336914
// MI455X (gfx1250) — compile-verified
//
#include <hip/hip_runtime.h>

// ---------------------------------------------------------------------------
// Neighbour-list build on MI455X (gfx1250, wave32, WMMA).
//
// r2(i,j) = n_i + n_j - 2*dot(x_i,x_j) is produced ENTIRELY by one
// V_WMMA_F32_16X16X4_F32 per 16x16 tile:
//     A row i  = ( x_i,  y_i,  z_i, 1.0 )
//     B col j  = (-2x_j,-2y_j,-2z_j, n_j)      (pre-scaled in LDS)
//     C        = n_i broadcast per row (matches C VGPR layout exactly)
//     D        = r2 tile, no post-WMMA arithmetic needed.
// Empty-tile fast path: per-lane fmin tree over the 8 D components + one
// ballot ("any pair in tile under cutoff") -> uniform branch; the per-row
// ballots + ordered drain only run on hit tiles.
// One wave walks all j-tiles in order so ascending-index compaction matches
// jnp.argwhere semantics; the diagonal is cleared scalar-side at j0==i0.
// ---------------------------------------------------------------------------

#define N_POINTS 8192
#define MAXNB    32
#define CUTOFF2  25.0f

typedef __attribute__((ext_vector_type(2))) float v2f;
typedef __attribute__((ext_vector_type(8))) float v8f;

// Pass 1: zero cell_indices region + max-neighbour slot (harness poisons).
__global__ __launch_bounds__(256) void nb_init_kernel(int* __restrict__ out) {
  const int t      = blockIdx.x * blockDim.x + threadIdx.x;
  const int stride = gridDim.x * blockDim.x;
  const int zbase  = N_POINTS * MAXNB;          // after to_idx
  const int ztot   = N_POINTS * MAXNB * 3 + 1;  // cell_indices + max slot
  for (int i = t; i < ztot; i += stride) out[zbase + i] = 0;
}

// Pass 2: 64 blocks x 8 waves; each wave owns a 16-row i-tile and scans all
// 512 j-tiles with one f32 WMMA per tile.
__global__ __launch_bounds__(256) void nb_neigh_kernel(
    const float* __restrict__ pos, int* __restrict__ out_idx,
    int* __restrict__ out_max) {
  // Staged B-side data, pre-scaled: 4 * 32KB = 128KB (<= 320KB/WGP).
  __shared__ float m2x[N_POINTS];   // -2*x
  __shared__ float m2y[N_POINTS];   // -2*y
  __shared__ float m2z[N_POINTS];   // -2*z
  __shared__ float sn[N_POINTS];    // x^2+y^2+z^2

  const int tid = threadIdx.x;
  for (int i = tid; i < N_POINTS; i += 256) {
    const float x = pos[3 * i + 0];
    const float y = pos[3 * i + 1];
    const float z = pos[3 * i + 2];
    m2x[i] = -2.0f * x;
    m2y[i] = -2.0f * y;
    m2z[i] = -2.0f * z;
    sn[i]  = x * x + y * y + z * z;
  }
  __syncthreads();

  const int  lane   = tid & 31;
  const int  wave   = tid >> 5;
  const int  i0     = (blockIdx.x * 8 + wave) * 16;  // this wave's row tile
  const int  lo     = lane & 15;
  const bool hiHalf = lane >= 16;

  // A tile (16x4 f32): lanes 0-15 -> (K0,K1)=(x,y); lanes 16-31 -> (K2,K3)=(z,1).
  const int arow = i0 + lo;
  v2f A;
  A[0] = hiHalf ? pos[3 * arow + 2] : pos[3 * arow + 0];
  A[1] = hiHalf ? 1.0f : pos[3 * arow + 1];

  // C accumulator = row norms n_i; C VGPR v holds row v (lanes 0-15) and
  // row v+8 (lanes 16-31), so the same value across each half-wave.
  v8f Cn;
#pragma unroll
  for (int v = 0; v < 8; ++v) Cn[v] = sn[i0 + v + (hiHalf ? 8 : 0)];

  int cnt   = 0;  // capped slot count    (valid in lanes 0..15)
  int total = 0;  // uncapped hit count   (valid in lanes 0..15)
  const int myrow = i0 + lane;  // row owned by lanes 0..15 for compaction

#pragma unroll 2
  for (int j0 = 0; j0 < N_POINTS; j0 += 16) {
    // B tile (4x16 f32): lanes 0-15 -> (K0,K1)=(-2x,-2y);
    //                    lanes 16-31 -> (K2,K3)=(-2z, n_j).
    const int bcol = j0 + lo;
    v2f B;
    B[0] = hiHalf ? m2z[bcol] : m2x[bcol];
    B[1] = hiHalf ? sn[bcol] : m2y[bcol];

    // D = A*B + C = full r2 tile.  EXEC all-1s here (uniform control flow).
    v8f D = __builtin_amdgcn_wmma_f32_16x16x4_f32(
        /*neg_a=*/false, A, /*neg_b=*/false, B,
        /*c_mod=*/(short)0, Cn, /*reuse_a=*/false, /*reuse_b=*/false);

    // Per-lane min over the 8 D components covers all 256 tile entries
    // across the wave; one ballot decides "any hit" (uniform branch).
    const float mn = fminf(fminf(fminf(D[0], D[1]), fminf(D[2], D[3])),
                           fminf(fminf(D[4], D[5]), fminf(D[6], D[7])));
    if (__builtin_amdgcn_ballot_w32(mn <= CUTOFF2) != 0u) {
      // Scalar branch on a wave-uniform value: EXEC still all-1s here.
      unsigned int bal[8];
#pragma unroll
      for (int v = 0; v < 8; ++v)
        bal[v] = __builtin_amdgcn_ballot_w32(D[v] <= CUTOFF2);

      // Lanes 0..15 each own one row; drain this tile's 16-bit mask in order.
      if (lane < 16) {
        const unsigned int m  = bal[lane & 7];
        unsigned int       rm = (lane < 8) ? (m & 0xFFFFu) : (m >> 16);
        if (j0 == i0) rm &= ~(1u << lane);  // drop self (diagonal tile only)
        total += __builtin_popcount(rm);
        while (rm) {
          const int b = __builtin_ctz(rm);
          rm &= rm - 1u;
          if (cnt < MAXNB) {
            out_idx[myrow * MAXNB + cnt] = j0 + b;
            ++cnt;
          }
        }
      }
    }
  }

  if (lane < 16) {
    for (int c = cnt; c < MAXNB; ++c) out_idx[myrow * MAXNB + c] = -1;
    atomicMax(out_max, total);
  }
}

extern "C" void kernel_launch(void* const* d_in, const int* in_sizes, int n_in,
                              void* d_out, int out_size, void* d_ws,
                              size_t ws_size, hipStream_t stream) {
  (void)in_sizes; (void)n_in; (void)out_size; (void)d_ws; (void)ws_size;
  const float* pos = (const float*)d_in[0];  // [8192,3] f32
  // d_in[1] = max_neighbours scalar (compile-time 32 here)
  int* out = (int*)d_out;  // [to_idx | cell_indices | max] as int32 bits

  nb_init_kernel<<<64, 256, 0, stream>>>(out);
  nb_neigh_kernel<<<64, 256, 0, stream>>>(
      pos, out, out + N_POINTS * MAXNB * 4 /* = 1048576 */);
}